// PNAGIN_37580963840347
// MI455X (gfx1250) — compile-verified
//
#include <hip/hip_runtime.h>

#define N_NODES   50000
#define N_EDGES   800000
#define NUM_GRAPHS 64
#define F_DIM     64
#define H_DIM     64
#define N_LAYERS  3
#define N_CLASSES 16

typedef float v2f __attribute__((ext_vector_type(2)));
typedef float v8f __attribute__((ext_vector_type(8)));

// D = A(16x4) * B(4x16) + C, exact fp32 (V_WMMA_F32_16X16X4_F32)
__device__ __forceinline__ v8f wmma4(v2f a, v2f b, v8f c) {
  return __builtin_amdgcn_wmma_f32_16x16x4_f32(false, a, false, b, (short)0, c, false, false);
}

__device__ __forceinline__ v2f ld2(const float* p) { return *(const v2f*)p; }

// monotone float -> uint mapping so integer atomicMin/Max implement float min/max
__device__ __forceinline__ unsigned encF(float f) {
  unsigned u = __float_as_uint(f);
  return (u & 0x80000000u) ? ~u : (u | 0x80000000u);
}
__device__ __forceinline__ float decF(unsigned u) {
  unsigned b = (u & 0x80000000u) ? (u ^ 0x80000000u) : ~u;
  return __uint_as_float(b);
}

__global__ void k_zero(float* __restrict__ p, long n) {
  long i = (long)blockIdx.x * 256 + threadIdx.x;
  if (i < n) p[i] = 0.0f;
}

__global__ void k_reset_agg(float* __restrict__ s, float* __restrict__ q,
                            unsigned* __restrict__ mn, unsigned* __restrict__ mx, long n) {
  long i = (long)blockIdx.x * 256 + threadIdx.x;
  if (i < n) { s[i] = 0.0f; q[i] = 0.0f; mn[i] = 0xFFFFFFFFu; mx[i] = 0u; }
}

__global__ void k_degree(const int* __restrict__ dst, float* __restrict__ deg, int e) {
  int i = blockIdx.x * 256 + threadIdx.x;
  if (i < e) atomicAdd(&deg[dst[i]], 1.0f);
}

__global__ void k_avglog(const float* __restrict__ deg, float* __restrict__ avgs, int n) {
  __shared__ float red[256];
  int i = blockIdx.x * 256 + threadIdx.x;
  float v = (i < n) ? logf(deg[i] + 1.0f) : 0.0f;
  red[threadIdx.x] = v;
  __syncthreads();
  for (int s = 128; s > 0; s >>= 1) {
    if (threadIdx.x < s) red[threadIdx.x] += red[threadIdx.x + s];
    __syncthreads();
  }
  if (threadIdx.x == 0) atomicAdd(avgs, red[0]);
}

__global__ void k_ampatt(const float* __restrict__ deg, const float* __restrict__ avgs,
                         float* __restrict__ amp, float* __restrict__ att, int n) {
  int i = blockIdx.x * 256 + threadIdx.x;
  if (i >= n) return;
  float avg = avgs[0] / (float)n;
  float dr = deg[i];
  float d  = dr > 1.0f ? dr : 1.0f;
  float ld = logf(d + 1.0f);
  amp[i] = ld / avg;
  att[i] = avg / ld;
}

// ---- edge kernel: m = [x_dst | x_src] @ preW^T + preB, scatter-aggregate by dst ----
__global__ __launch_bounds__(256) void k_edge(
    const float* __restrict__ x, const int* __restrict__ src, const int* __restrict__ dst,
    const float* __restrict__ preW, const float* __restrict__ preB,
    float* __restrict__ aSum, float* __restrict__ aSq,
    unsigned* __restrict__ aMn, unsigned* __restrict__ aMx) {
  __shared__ float sW[64 * 128];   // preW [64 out][128 in], 32KB
  int tid = threadIdx.x;
  for (int i = tid; i < 64 * 128; i += 256) sW[i] = preW[i];
  __syncthreads();

  int wid = tid >> 5, lane = tid & 31;
  int g  = blockIdx.x * 8 + wid;          // 16-edge group
  int r  = lane & 15, kh = lane >> 4;
  long e = (long)g * 16 + r;
  if (e > (long)N_EDGES - 1) e = N_EDGES - 1;
  const float* pd = x + (long)dst[e] * 64;
  const float* ps = x + (long)src[e] * 64;

  v8f acc[4] = { {}, {}, {}, {} };

#define EDGE_KLOOP(APTR, KOFF)                                        \
  _Pragma("unroll 4")                                                 \
  for (int kb = 0; kb < 16; ++kb) {                                   \
    int fk = kb * 4 + 2 * kh;                                         \
    v2f a  = ld2((APTR) + fk);                                        \
    int kc = (KOFF) + fk;                                             \
    v2f b0 = ld2(&sW[(r     ) * 128 + kc]);                           \
    v2f b1 = ld2(&sW[(r + 16) * 128 + kc]);                           \
    v2f b2 = ld2(&sW[(r + 32) * 128 + kc]);                           \
    v2f b3 = ld2(&sW[(r + 48) * 128 + kc]);                           \
    acc[0] = wmma4(a, b0, acc[0]);                                    \
    acc[1] = wmma4(a, b1, acc[1]);                                    \
    acc[2] = wmma4(a, b2, acc[2]);                                    \
    acc[3] = wmma4(a, b3, acc[3]);                                    \
  }

  EDGE_KLOOP(pd, 0)     // features 0..63  from x[dst]
  EDGE_KLOOP(ps, 64)    // features 64..127 from x[src]
#undef EDGE_KLOOP

  // epilogue: D[v] = m[edge = g*16 + v + 8*kh][col = nt*16 + r]
  int nodev[8];
#pragma unroll
  for (int v = 0; v < 8; ++v) {
    long er = (long)g * 16 + v + 8 * kh;
    nodev[v] = dst[er > (long)N_EDGES - 1 ? N_EDGES - 1 : er];
  }
#pragma unroll
  for (int nt = 0; nt < 4; ++nt) {
    int n = nt * 16 + r;
    float bn = preB[n];
#pragma unroll
    for (int v = 0; v < 8; ++v) {
      long er = (long)g * 16 + v + 8 * kh;
      if (er < N_EDGES) {
        float m = acc[nt][v] + bn;
        long o = (long)nodev[v] * 64 + n;
        atomicAdd(&aSum[o], m);
        atomicAdd(&aSq[o], m * m);
        atomicMin(&aMn[o], encF(m));
        atomicMax(&aMx[o], encF(m));
      }
    }
  }
}

// ---- finalize: in-place mean/std + decoded masked min/max ----
__global__ void k_finalize(float* __restrict__ aSum, float* __restrict__ aSq,
                           unsigned* __restrict__ aMn, unsigned* __restrict__ aMx,
                           const float* __restrict__ deg, long total) {
  long i = (long)blockIdx.x * 256 + threadIdx.x;
  if (i >= total) return;
  float dr = deg[i >> 6];
  float d  = dr > 1.0f ? dr : 1.0f;
  float s1 = aSum[i], s2 = aSq[i];
  float mean = s1 / d;
  float var  = s2 / d - mean * mean;
  float sd   = sqrtf((var > 0.0f ? var : 0.0f) + 1e-5f);
  bool  has  = dr > 0.0f;
  float mn = has ? decF(aMn[i]) : 0.0f;
  float mx = has ? decF(aMx[i]) : 0.0f;
  aSum[i] = mean;
  aSq[i]  = sd;
  ((float*)aMn)[i] = mn;
  ((float*)aMx)[i] = mx;
}

// ---- node kernel: post GEMM (split 13F trick) + lin GEMM + ReLU ----
__global__ __launch_bounds__(256) void k_node(
    const float* __restrict__ xin,
    const float* __restrict__ aMean, const float* __restrict__ aMn,
    const float* __restrict__ aMx,  const float* __restrict__ aStd,
    const float* __restrict__ amp,  const float* __restrict__ att,
    const float* __restrict__ postW, const float* __restrict__ postB,
    const float* __restrict__ linW,  const float* __restrict__ linB,
    float* __restrict__ xout, int ngroups) {
  __shared__ float sLin[64 * 64];       // 16KB
  __shared__ float sOut[8][16 * 64];    // 32KB staging of post result per wave
  int tid = threadIdx.x;
  for (int i = tid; i < 64 * 64; i += 256) sLin[i] = linW[i];
  __syncthreads();

  int wid = tid >> 5, lane = tid & 31;
  int g  = blockIdx.x * 8 + wid;
  int gi = g < ngroups ? g : ngroups - 1;
  int r  = lane & 15, kh = lane >> 4;
  long nrow = (long)gi * 16 + r;        // ngroups*16 == N exactly
  const float* xrow = xin   + nrow * 64;
  const float* rMe  = aMean + nrow * 64;
  const float* rMn  = aMn   + nrow * 64;
  const float* rMx  = aMx   + nrow * 64;
  const float* rSd  = aStd  + nrow * 64;

  float ampv[8], attv[8];
#pragma unroll
  for (int v = 0; v < 8; ++v) {
    long nn = (long)gi * 16 + v + 8 * kh;
    ampv[v] = amp[nn];
    attv[v] = att[nn];
  }

#pragma unroll 1
  for (int nt = 0; nt < 4; ++nt) {
    int n = nt * 16 + r;
    const float* wrow = postW + (long)n * 832;
    v8f aX = {}, aM = {}, aP = {}, aT = {};
#pragma unroll 4
    for (int kb = 0; kb < 16; ++kb) {   // x block, post cols 0..63
      int fk = kb * 4 + 2 * kh;
      aX = wmma4(ld2(xrow + fk), ld2(wrow + fk), aX);
    }
#define AGG_KLOOP(PTR, COFF)                                          \
    _Pragma("unroll 4")                                               \
    for (int kb = 0; kb < 16; ++kb) {                                 \
      int fk = kb * 4 + 2 * kh;                                       \
      v2f a = ld2((PTR) + fk);                                        \
      aM = wmma4(a, ld2(wrow +  64 + (COFF) + fk), aM);               \
      aP = wmma4(a, ld2(wrow + 320 + (COFF) + fk), aP);               \
      aT = wmma4(a, ld2(wrow + 576 + (COFF) + fk), aT);               \
    }
    AGG_KLOOP(rMe, 0)
    AGG_KLOOP(rMn, 64)
    AGG_KLOOP(rMx, 128)
    AGG_KLOOP(rSd, 192)
#undef AGG_KLOOP
    float bn = postB[n];
#pragma unroll
    for (int v = 0; v < 8; ++v) {
      int row = v + 8 * kh;
      sOut[wid][row * 64 + n] = aX[v] + aM[v] + ampv[v] * aP[v] + attv[v] * aT[v] + bn;
    }
  }
  __syncthreads();

#pragma unroll 1
  for (int nt = 0; nt < 4; ++nt) {
    int n = nt * 16 + r;
    v8f acc = {};
#pragma unroll 4
    for (int kb = 0; kb < 16; ++kb) {
      int fk = kb * 4 + 2 * kh;
      acc = wmma4(ld2(&sOut[wid][r * 64 + fk]), ld2(&sLin[n * 64 + fk]), acc);
    }
    float bn = linB[n];
#pragma unroll
    for (int v = 0; v < 8; ++v) {
      int row = v + 8 * kh;
      if (g < ngroups) {
        float o = acc[v] + bn;
        xout[((long)g * 16 + row) * 64 + n] = o > 0.0f ? o : 0.0f;
      }
    }
  }
}

__global__ void k_pool(const float* __restrict__ xf, const int* __restrict__ batch,
                       float* __restrict__ psum, float* __restrict__ pcnt, long total) {
  long i = (long)blockIdx.x * 256 + threadIdx.x;
  if (i >= total) return;
  int nIdx = (int)(i >> 6), f = (int)(i & 63);
  int b = batch[nIdx];
  atomicAdd(&psum[b * 64 + f], xf[i]);
  if (f == 0) atomicAdd(&pcnt[b], 1.0f);
}

__global__ __launch_bounds__(64) void k_final(
    const float* __restrict__ psum, const float* __restrict__ pcnt,
    const float* __restrict__ w1, const float* __restrict__ b1,
    const float* __restrict__ w2, const float* __restrict__ b2, float* __restrict__ out) {
  __shared__ float sg[64], sh[64];
  int g = blockIdx.x, t = threadIdx.x;
  float c = pcnt[g];
  c = c > 1.0f ? c : 1.0f;
  sg[t] = psum[g * 64 + t] / c;
  __syncthreads();
  float a = b1[t];
  for (int f = 0; f < 64; ++f) a += w1[t * 64 + f] * sg[f];
  sh[t] = a > 0.0f ? a : 0.0f;
  __syncthreads();
  if (t < N_CLASSES) {
    float o = b2[t];
    for (int h = 0; h < 64; ++h) o += w2[t * 64 + h] * sh[h];
    out[g * N_CLASSES + t] = o;
  }
}

extern "C" void kernel_launch(void* const* d_in, const int* in_sizes, int n_in,
                              void* d_out, int out_size, void* d_ws, size_t ws_size,
                              hipStream_t stream) {
  const float* x     = (const float*)d_in[0];
  const int*   eidx  = (const int*)  d_in[1];
  const int*   batch = (const int*)  d_in[2];
  const float* preW  = (const float*)d_in[3];
  const float* preB  = (const float*)d_in[4];
  const float* postW = (const float*)d_in[5];
  const float* postB = (const float*)d_in[6];
  const float* linW  = (const float*)d_in[7];
  const float* linB  = (const float*)d_in[8];
  const float* w1    = (const float*)d_in[9];
  const float* b1    = (const float*)d_in[10];
  const float* w2    = (const float*)d_in[11];
  const float* b2    = (const float*)d_in[12];
  float* out = (float*)d_out;

  const int* srcI = eidx;
  const int* dstI = eidx + N_EDGES;

  float* wsf = (float*)d_ws;
  size_t o = 0;
  float* deg  = wsf + o; o += N_NODES;
  float* avgs = wsf + o; o += 64;                    // padded scalar
  float* ampA = wsf + o; o += N_NODES;
  float* attA = wsf + o; o += N_NODES;
  float* aSum = wsf + o; o += (size_t)N_NODES * 64;
  float* aSq  = wsf + o; o += (size_t)N_NODES * 64;
  unsigned* aMn = (unsigned*)(wsf + o); o += (size_t)N_NODES * 64;
  unsigned* aMx = (unsigned*)(wsf + o); o += (size_t)N_NODES * 64;
  float* xa   = wsf + o; o += (size_t)N_NODES * 64;
  float* xb   = wsf + o; o += (size_t)N_NODES * 64;
  float* psum = wsf + o; o += 64 * 64;
  float* pcnt = wsf + o; o += 64;                    // contiguous after psum

  const long NF = (long)N_NODES * 64;
  dim3 b256(256);

  k_zero<<<dim3((N_NODES + 255) / 256), b256, 0, stream>>>(deg, N_NODES);
  k_zero<<<dim3(1), b256, 0, stream>>>(avgs, 1);
  k_zero<<<dim3((64 * 64 + 64 + 255) / 256), b256, 0, stream>>>(psum, 64 * 64 + 64);
  k_degree<<<dim3((N_EDGES + 255) / 256), b256, 0, stream>>>(dstI, deg, N_EDGES);
  k_avglog<<<dim3((N_NODES + 255) / 256), b256, 0, stream>>>(deg, avgs, N_NODES);
  k_ampatt<<<dim3((N_NODES + 255) / 256), b256, 0, stream>>>(deg, avgs, ampA, attA, N_NODES);

  const float* xi = x;
  float* xo = xa;
  int eBlocks = (N_EDGES / 16 + 7) / 8;   // 6250
  int ngroups = N_NODES / 16;             // 3125
  int nBlocks = (ngroups + 7) / 8;        // 391
  for (int l = 0; l < N_LAYERS; ++l) {
    k_reset_agg<<<dim3((int)((NF + 255) / 256)), b256, 0, stream>>>(aSum, aSq, aMn, aMx, NF);
    k_edge<<<dim3(eBlocks), b256, 0, stream>>>(xi, srcI, dstI,
        preW + (size_t)l * 64 * 128, preB + (size_t)l * 64, aSum, aSq, aMn, aMx);
    k_finalize<<<dim3((int)((NF + 255) / 256)), b256, 0, stream>>>(aSum, aSq, aMn, aMx, deg, NF);
    k_node<<<dim3(nBlocks), b256, 0, stream>>>(xi, aSum, (const float*)aMn, (const float*)aMx, aSq,
        ampA, attA, postW + (size_t)l * 64 * 832, postB + (size_t)l * 64,
        linW + (size_t)l * 64 * 64, linB + (size_t)l * 64, xo, ngroups);
    xi = xo;
    xo = (xo == xa) ? xb : xa;
  }
  k_pool<<<dim3((int)((NF + 255) / 256)), b256, 0, stream>>>(xi, batch, psum, pcnt, NF);
  k_final<<<dim3(NUM_GRAPHS), dim3(64), 0, stream>>>(psum, pcnt, w1, b1, w2, b2, out);
}